// RNNTLoss_62775241999018
// MI455X (gfx1250) — compile-verified
//
#include <hip/hip_runtime.h>
#include <cmath>
#include <stdint.h>

#define B_  16
#define T_  160
#define U_  80
#define U1  81
#define V_  512

typedef float v8f __attribute__((ext_vector_type(8)));
typedef float v2f __attribute__((ext_vector_type(2)));
typedef int   v4i __attribute__((__vector_size__(4 * sizeof(int))));

typedef __attribute__((address_space(1))) v4i as1_v4i;   // global 16B vector
typedef __attribute__((address_space(3))) v4i as3_v4i;   // LDS    16B vector

#if defined(__has_builtin)
#if __has_builtin(__builtin_amdgcn_global_load_async_to_lds_b128)
#define HAVE_ASYNC_LDS 1
#endif
#endif

static __device__ __forceinline__ void wait_asynccnt0() {
#if defined(__has_builtin)
#if __has_builtin(__builtin_amdgcn_s_wait_asynccnt)
    __builtin_amdgcn_s_wait_asynccnt(0);
#else
    asm volatile("s_wait_asynccnt 0x0" ::: "memory");
#endif
#else
    asm volatile("s_wait_asynccnt 0x0" ::: "memory");
#endif
}

// ---------------------------------------------------------------------------
// Kernel A: per-row softmax stats for 16 rows of 512 f32 per block.
//   blank_lp[g] = x[511] - logZ,  lab_lp[g] = x[targets[b,u]] - logZ
// 32 KB tile staged into LDS via GLOBAL_LOAD_ASYNC_TO_LDS_B128 (ASYNCcnt),
// sum-of-exp via chained V_WMMA_F32_16X16X4_F32 (B = ones), full f32.
// ---------------------------------------------------------------------------
__global__ __launch_bounds__(256) void rnnt_softmax_stats(
    const float* __restrict__ logits, const int* __restrict__ targets,
    float* __restrict__ blank_out, float* __restrict__ lab_out)
{
    constexpr int ROWPAD = 516;                 // 512 + 4: bank stride 4, b128-aligned
    __shared__ float smem[16 * ROWPAD];
    __shared__ float rowmax[16];
    __shared__ float wsum[16][8];

    const int tid  = threadIdx.x;
    const int lane = tid & 31;
    const int w    = tid >> 5;
    const long long group = blockIdx.x;         // 16 consecutive (b,t,u) rows
    const float* src = logits + group * (16LL * V_);

    // ---- stage 16x512 into LDS (padded rows) ----
#ifdef HAVE_ASYNC_LDS
    #pragma unroll
    for (int i = 0; i < 8; ++i) {
        int f = i * 1024 + tid * 4;
        int r = f >> 9;
        int c = f & (V_ - 1);
        as1_v4i* gp = (as1_v4i*)(uintptr_t)(src + f);
        as3_v4i* lp = (as3_v4i*)(uint32_t)(uintptr_t)(&smem[r * ROWPAD + c]);
        __builtin_amdgcn_global_load_async_to_lds_b128(gp, lp, 0, 0);
    }
    wait_asynccnt0();
#else
    #pragma unroll
    for (int i = 0; i < 8; ++i) {
        int f = i * 1024 + tid * 4;
        int r = f >> 9;
        int c = f & (V_ - 1);
        const float4 v = *reinterpret_cast<const float4*>(src + f);
        *reinterpret_cast<float4*>(&smem[r * ROWPAD + c]) = v;
    }
#endif
    __syncthreads();

    // ---- per-row max: wave w handles rows 2w, 2w+1 ----
    #pragma unroll
    for (int rr = 0; rr < 2; ++rr) {
        int row = w * 2 + rr;
        float m = -3.0e38f;
        #pragma unroll
        for (int j = 0; j < 16; ++j)
            m = fmaxf(m, smem[row * ROWPAD + lane + j * 32]);
        #pragma unroll
        for (int off = 16; off >= 1; off >>= 1)
            m = fmaxf(m, __shfl_xor(m, off, 32));
        if (lane == 0) rowmax[row] = m;
    }
    __syncthreads();

    // ---- sum of exp(x - max) via chained V_WMMA_F32_16X16X4_F32 ----
    // A layout (32-bit 16x4): lane l<16 -> row l, K={0,1}; lane l+16 -> row l, K={2,3}
    // B = ones => D[m][n] = sum_k A[m][k]; wave w covers K-slice [64w, 64w+64)
    const int   arow = lane & 15;
    const int   kadd = (lane >> 4) << 1;
    const float mx   = rowmax[arow];
    const float* rp  = &smem[arow * ROWPAD];

    v8f c = {0.f, 0.f, 0.f, 0.f, 0.f, 0.f, 0.f, 0.f};
    v2f bones; bones[0] = 1.0f; bones[1] = 1.0f;

    #pragma unroll
    for (int it = 0; it < 16; ++it) {
        int k0 = w * 64 + it * 4 + kadd;
        v2f a;
        a[0] = __expf(rp[k0]     - mx);   // arg <= 0: raw v_exp path is exact enough
        a[1] = __expf(rp[k0 + 1] - mx);
        c = __builtin_amdgcn_wmma_f32_16x16x4_f32(
                /*neg_a=*/false, a, /*neg_b=*/false, bones,
                /*c_mod=*/(short)0, c, /*reuse_a=*/false, /*reuse_b=*/false);
    }

    // D layout: VGPR r, lanes 0-15 -> M=r; lanes 16-31 -> M=8+r (all N identical)
    float partial = 0.0f;
    #pragma unroll
    for (int m = 0; m < 16; ++m) {
        float v = __shfl(c[m & 7], (m >> 3) * 16, 32);
        if (lane == m) partial = v;
    }
    if (lane < 16) wsum[lane][w] = partial;
    __syncthreads();

    // ---- combine 8 wave partials, emit blank/label log-probs ----
    if (tid < 16) {
        float tot = 0.0f;
        #pragma unroll
        for (int ww = 0; ww < 8; ++ww) tot += wsum[tid][ww];
        float logZ = rowmax[tid] + logf(tot);   // precise log on the cold path

        long long g  = group * 16 + tid;
        int       u  = (int)(g % U1);
        long long bt = g / U1;                  // b*T + t
        int       b  = (int)(bt / T_);

        blank_out[g] = smem[tid * ROWPAD + (V_ - 1)] - logZ;
        float labv = 0.0f;
        if (u < U_) {
            int tgt = targets[b * U_ + u];
            labv = smem[tid * ROWPAD + tgt] - logZ;
        }
        lab_out[g] = labv;
    }
}

// ---------------------------------------------------------------------------
// Kernel B: alpha DP over the (T x U+1) lattice, anti-diagonal wavefront.
// One block per batch element; alpha lives in LDS (160*81*4 = 51.8 KB).
// alpha[t][u] = logaddexp(alpha[t-1][u]+blank[t-1][u], alpha[t][u-1]+lab[t][u-1])
// ---------------------------------------------------------------------------
__global__ __launch_bounds__(128) void rnnt_alpha(
    const float* __restrict__ blank_lp, const float* __restrict__ lab_lp,
    const int* __restrict__ logit_lengths, const int* __restrict__ target_lengths,
    float* __restrict__ ll_out)
{
    __shared__ float alpha[T_ * U1];
    const int b   = blockIdx.x;
    const int tid = threadIdx.x;
    const float* bl = blank_lp + (long long)b * T_ * U1;
    const float* la = lab_lp   + (long long)b * T_ * U1;

    if (tid == 0) alpha[0] = 0.0f;
    __syncthreads();

    for (int d = 1; d < T_ + U_; ++d) {
        int tlo = d - U_ > 0 ? d - U_ : 0;
        int thi = d < T_ - 1 ? d : T_ - 1;
        int t = tlo + tid;
        if (t <= thi) {
            int u = d - t;
            float val;
            if (t == 0) {
                val = alpha[u - 1] + la[u - 1];
            } else if (u == 0) {
                val = alpha[(t - 1) * U1] + bl[(t - 1) * U1];
            } else {
                float x1 = alpha[(t - 1) * U1 + u] + bl[(t - 1) * U1 + u];
                float x2 = alpha[t * U1 + (u - 1)] + la[t * U1 + (u - 1)];
                float m  = fmaxf(x1, x2);
                val = m + log1pf(expf(-fabsf(x1 - x2)));
            }
            alpha[t * U1 + u] = val;
        }
        __syncthreads();
    }

    if (tid == 0) {
        int ti = logit_lengths[b] - 1;
        int ui = target_lengths[b];
        ll_out[b] = alpha[ti * U1 + ui] + bl[ti * U1 + ui];
    }
}

// ---------------------------------------------------------------------------
// Kernel C: deterministic scalar mean of -log_like (no float atomics).
// ---------------------------------------------------------------------------
__global__ void rnnt_finalize(const float* __restrict__ ll, float* __restrict__ out)
{
    if (threadIdx.x == 0) {
        float s = 0.0f;
        #pragma unroll
        for (int i = 0; i < B_; ++i) s += ll[i];
        out[0] = -s / (float)B_;
    }
}

extern "C" void kernel_launch(void* const* d_in, const int* in_sizes, int n_in,
                              void* d_out, int out_size, void* d_ws, size_t ws_size,
                              hipStream_t stream) {
    (void)in_sizes; (void)n_in; (void)out_size; (void)ws_size;
    const float* logits  = (const float*)d_in[0];
    const int*   targets = (const int*)d_in[1];
    const int*   loglen  = (const int*)d_in[2];
    const int*   tgtlen  = (const int*)d_in[3];
    float* out = (float*)d_out;

    float* ws    = (float*)d_ws;
    float* blank = ws;                                   // B*T*(U+1) floats
    float* lab   = ws + (size_t)B_ * T_ * U1;            // B*T*(U+1) floats
    float* ll    = lab + (size_t)B_ * T_ * U1;           // B floats

    const int ngroups = (B_ * T_ * U1) / 16;             // 12960
    rnnt_softmax_stats<<<ngroups, 256, 0, stream>>>(logits, targets, blank, lab);
    rnnt_alpha<<<B_, 128, 0, stream>>>(blank, lab, loglen, tgtlen, ll);
    rnnt_finalize<<<1, 32, 0, stream>>>(ll, out);
}